// DimGraphConv_43989055046247
// MI455X (gfx1250) — compile-verified
//
#include <hip/hip_runtime.h>

// GraphSAGE-style conv: out = x@Wself^T + mean_agg(x[src]->dst)@Wneigh^T + bias
// Fused: aggregate raw features first (linear op commutes with GEMM), then one
// WMMA f32 GEMM over nodes. N=100000, E=1600000, D=64.

#define N_NODES 100000
#define N_EDGES 1600000
#define DFEAT   64
#define NTILES  (N_NODES / 16)   // 6250, exact

typedef __attribute__((ext_vector_type(2))) float v2f;
typedef __attribute__((ext_vector_type(8))) float v8f;

// ---------------- Kernel 1: zero workspace (agg [N*64] + deg [N]) ----------
__global__ __launch_bounds__(256) void dgc_zero(float4* __restrict__ agg4,
                                                float* __restrict__ deg) {
    int i = blockIdx.x * blockDim.x + threadIdx.x;
    if (i < N_NODES * 16) agg4[i] = make_float4(0.f, 0.f, 0.f, 0.f);
    if (i < N_NODES)      deg[i] = 0.f;
}

// ---------------- Kernel 2: edge scatter-add aggregation -------------------
// One thread per (edge, 4-feature group): float4 gather from x[src], 4 f32
// atomics into agg[dst]. x (25.6MB) and agg (25.6MB) both L2-resident.
__global__ __launch_bounds__(256) void dgc_scatter(const float* __restrict__ x,
                                                   const long long* __restrict__ ei,
                                                   float* __restrict__ agg,
                                                   float* __restrict__ deg) {
    int t = blockIdx.x * blockDim.x + threadIdx.x;
    int e  = t >> 4;
    int f4 = t & 15;
    if (e >= N_EDGES) return;
    int src = (int)ei[e];             // row (source)
    int dst = (int)ei[N_EDGES + e];   // col (target)
    const float4 v = *(const float4*)(x + src * DFEAT + f4 * 4);
    float* a = agg + dst * DFEAT + f4 * 4;
    atomicAdd(a + 0, v.x);
    atomicAdd(a + 1, v.y);
    atomicAdd(a + 2, v.z);
    atomicAdd(a + 3, v.w);
    if (f4 == 0) atomicAdd(deg + dst, 1.0f);
}

// ---------------- Kernel 3: fused dual-GEMM via V_WMMA_F32_16X16X4_F32 -----
// One wave handles a 16-node x 64-out tile: 4 f32 accumulators, A-fragment
// (x row / scaled agg row) reused across the 4 output-column tiles.
// f32 A(16x4)/B(4x16) lane layout: lane L holds K = 2*(L>>4)+{0,1}  -> float2
// C/D layout: VGPR r holds M = r + 8*(L>>4), N = L&15.
__global__ __launch_bounds__(256) void dgc_gemm(const float* __restrict__ x,
                                                const float* __restrict__ agg,
                                                const float* __restrict__ deg,
                                                const float* __restrict__ Wself,
                                                const float* __restrict__ Wneigh,
                                                const float* __restrict__ bias,
                                                float* __restrict__ out) {
    const int lane = threadIdx.x & 31;
    const int wave = (blockIdx.x * blockDim.x + threadIdx.x) >> 5;  // uniform per wave
    if (wave >= NTILES) return;                                     // wave-uniform exit

    const int node0 = wave * 16;
    const int m     = lane & 15;        // A row within tile; also B/C column N
    const int kh    = (lane >> 4) * 2;  // K sub-offset within 4-group: 0 or 2
    const int arow  = node0 + m;

    // inverse in-degree for this lane's A row (mean aggregation)
    const float inv = 1.0f / fmaxf(deg[arow], 1.0f);

    v8f acc[4] = {};

#pragma unroll
    for (int k0 = 0; k0 < DFEAT; k0 += 4) {
        v2f ax = *(const v2f*)(x   + arow * DFEAT + k0 + kh);
        v2f aa = *(const v2f*)(agg + arow * DFEAT + k0 + kh);
        aa.x *= inv;
        aa.y *= inv;
#pragma unroll
        for (int t = 0; t < 4; ++t) {
            const int n = t * 16 + m;   // output feature = B row in W[Dout][Din]
            v2f bs = *(const v2f*)(Wself  + n * DFEAT + k0 + kh);
            v2f bn = *(const v2f*)(Wneigh + n * DFEAT + k0 + kh);
            acc[t] = __builtin_amdgcn_wmma_f32_16x16x4_f32(
                false, ax, false, bs, (short)0, acc[t], false, false);
            acc[t] = __builtin_amdgcn_wmma_f32_16x16x4_f32(
                false, aa, false, bn, (short)0, acc[t], false, false);
        }
    }

    const int h = lane >> 4;
#pragma unroll
    for (int t = 0; t < 4; ++t) {
        const float b = bias[t * 16 + m];
#pragma unroll
        for (int r = 0; r < 8; ++r) {
            const int node = node0 + r + 8 * h;
            out[node * DFEAT + t * 16 + m] = acc[t][r] + b;
        }
    }
}

extern "C" void kernel_launch(void* const* d_in, const int* in_sizes, int n_in,
                              void* d_out, int out_size, void* d_ws, size_t ws_size,
                              hipStream_t stream) {
    const float*     x      = (const float*)d_in[0];
    const long long* ei     = (const long long*)d_in[1];   // int64 [2, E]
    const float*     Wself  = (const float*)d_in[2];
    const float*     Wneigh = (const float*)d_in[3];
    const float*     bias   = (const float*)d_in[4];
    float*           out    = (float*)d_out;

    float* agg = (float*)d_ws;                 // N*64 floats = 25.6 MB
    float* deg = agg + (size_t)N_NODES * DFEAT; // N floats

    // 1) zero accumulators (ws is poisoned; must re-zero every call)
    {
        int threads = N_NODES * 16;  // float4 granularity over agg; covers deg too
        dgc_zero<<<(threads + 255) / 256, 256, 0, stream>>>((float4*)agg, deg);
    }
    // 2) edge aggregation: agg[dst] += x[src]; deg[dst] += 1
    {
        int threads = N_EDGES * 16;
        dgc_scatter<<<(threads + 255) / 256, 256, 0, stream>>>(x, ei, agg, deg);
    }
    // 3) fused node GEMM with f32 WMMA
    {
        int threads = NTILES * 32;   // one wave32 per 16-node tile
        dgc_gemm<<<(threads + 255) / 256, 256, 0, stream>>>(x, agg, deg, Wself,
                                                            Wneigh, bias, out);
    }
}